// AttentionBlock_34797825032198
// MI455X (gfx1250) — compile-verified
//
#include <hip/hip_runtime.h>
#include <hip/hip_bf16.h>
#include <math.h>

typedef __attribute__((ext_vector_type(2))) float v2f;
typedef __attribute__((ext_vector_type(4))) float v4f;
typedef __attribute__((ext_vector_type(8))) float v8f;

#define BDIM   64      // spatial H=W
#define CCH    128     // channels
#define NHEADS 4
#define DH     32      // head dim
#define KW     7
#define HALO   14      // 8 + 6
#define NPIX   8192    // 2*64*64

// ---------------------------------------------------------------------------
// NCHW -> NHWC transpose
// ---------------------------------------------------------------------------
__global__ __launch_bounds__(256) void nchw_to_nhwc(const float* __restrict__ x,
                                                    float* __restrict__ y) {
    int t = blockIdx.x * blockDim.x + threadIdx.x;
    if (t >= 2 * 128 * 4096) return;
    int b  = t >> 19;          // 128*4096 = 2^19
    int c  = (t >> 12) & 127;
    int ij = t & 4095;
    y[(b * 4096 + ij) * CCH + c] = x[t];
}

// ---------------------------------------------------------------------------
// out[P x O] = A[P x 128] @ W[O x 128]^T + bias ; cols < scale_cols get *scale
// One wave computes a 16x32 tile using V_WMMA_F32_16X16X4_F32 (f32 matrix op).
// A-frag: lanes 0-15 hold K=kk,kk+1 ; lanes 16-31 hold K=kk+2,kk+3 (per ISA).
// ---------------------------------------------------------------------------
__global__ __launch_bounds__(256) void gemm_bias_wmma(
    const float* __restrict__ A, const float* __restrict__ W,
    const float* __restrict__ bias, float* __restrict__ out,
    int P, int O, int scale_cols, float scale)
{
    int gt   = blockIdx.x * blockDim.x + threadIdx.x;
    int wid  = gt >> 5;
    int lane = threadIdx.x & 31;
    int tx   = O >> 5;                 // 32 output cols per wave
    int tm   = wid / tx, tn = wid % tx;
    int m0   = tm << 4, n0 = tn << 5;
    if (m0 >= P) return;               // wave-uniform: EXEC stays all-1 for WMMA

    int l16 = lane & 15;
    int hi  = lane >> 4;               // 0: K pair {0,1}, 1: K pair {2,3}

    const float* arow = A + (size_t)(m0 + l16) * 128;
    const float* w0   = W + (size_t)(n0 + l16) * 128;
    const float* w1   = W + (size_t)(n0 + 16 + l16) * 128;

    v8f acc0 = {0.f,0.f,0.f,0.f,0.f,0.f,0.f,0.f};
    v8f acc1 = {0.f,0.f,0.f,0.f,0.f,0.f,0.f,0.f};

    #pragma unroll 8
    for (int kk = 0; kk < 128; kk += 4) {
        int kb = kk + hi * 2;
        v2f a  = *(const v2f*)(arow + kb);
        v2f b0 = *(const v2f*)(w0 + kb);
        v2f b1 = *(const v2f*)(w1 + kb);
        acc0 = __builtin_amdgcn_wmma_f32_16x16x4_f32(false, a, false, b0,
                                                     (short)0, acc0, false, false);
        acc1 = __builtin_amdgcn_wmma_f32_16x16x4_f32(false, a, false, b1,
                                                     (short)0, acc1, false, false);
    }

    int n_a = n0 + l16;
    int n_b = n0 + 16 + l16;
    float bia = bias[n_a], bib = bias[n_b];
    float sa  = (n_a < scale_cols) ? scale : 1.0f;
    float sb  = (n_b < scale_cols) ? scale : 1.0f;
    #pragma unroll
    for (int v = 0; v < 8; ++v) {
        int row = m0 + v + hi * 8;     // C/D layout: VGPR v -> M = v + 8*hi
        out[(size_t)row * O + n_a] = (acc0[v] + bia) * sa;
        out[(size_t)row * O + n_b] = (acc1[v] + bib) * sb;
    }
}

// ---------------------------------------------------------------------------
// Neighborhood attention: one block = (batch, head, 8x8 query tile).
// K/V for the 14x14 halo staged in LDS as [d][pixel] (conflict-friendly),
// q in LDS; per-thread fully-unrolled 7x7 scores + softmax + AV.
// ---------------------------------------------------------------------------
__global__ __launch_bounds__(64) void natten_kernel(
    const float* __restrict__ qkv,   // [NPIX][384] : q|k|v, q pre-scaled
    const float* __restrict__ rpb,   // [4][13][13]
    float* __restrict__ out)         // [NPIX][128]
{
    __shared__ float Klds[DH * 196];
    __shared__ float Vlds[DH * 196];
    __shared__ float Qlds[64 * 33];

    int bx   = blockIdx.x;
    int tj   = bx & 7;
    int ti   = (bx >> 3) & 7;
    int head = (bx >> 6) & 3;
    int b    = bx >> 8;
    int i0 = ti * 8, j0 = tj * 8;
    int r0 = min(max(i0 - 3, 0), 50);   // halo origin (covers all clamped windows)
    int c0 = min(max(j0 - 3, 0), 50);
    int tid = threadIdx.x;

    // cooperative halo K/V load (coalesced over d)
    for (int e = tid; e < 196 * DH; e += 64) {
        int px = e >> 5;                 // halo pixel 0..195
        int d  = e & 31;
        int gi = r0 + px / HALO;
        int gj = c0 + px % HALO;
        int gp = (b * BDIM + gi) * BDIM + gj;
        Klds[d * 196 + px] = qkv[(size_t)gp * 384 + 128 + head * DH + d];
        Vlds[d * 196 + px] = qkv[(size_t)gp * 384 + 256 + head * DH + d];
    }
    int li = tid >> 3, lj = tid & 7;
    int i = i0 + li, j = j0 + lj;
    int p = (b * BDIM + i) * BDIM + j;
    for (int d = 0; d < DH; ++d)
        Qlds[tid * 33 + d] = qkv[(size_t)p * 384 + head * DH + d];
    __syncthreads();

    int si = min(max(i - 3, 0), BDIM - KW);
    int sj = min(max(j - 3, 0), BDIM - KW);
    int base = (si - r0) * HALO + (sj - c0);
    int boi = (KW - 1) - (i - si);
    int boj = (KW - 1) - (j - sj);
    const float* rp = rpb + head * 169;

    float sc[49];
    #pragma unroll
    for (int pp = 0; pp < 7; ++pp)
        #pragma unroll
        for (int qq = 0; qq < 7; ++qq)
            sc[pp * 7 + qq] = rp[(pp + boi) * 13 + (qq + boj)];

    for (int d = 0; d < DH; ++d) {
        float qd = Qlds[tid * 33 + d];
        const float* kr = &Klds[d * 196 + base];
        #pragma unroll
        for (int pp = 0; pp < 7; ++pp)
            #pragma unroll
            for (int qq = 0; qq < 7; ++qq)
                sc[pp * 7 + qq] = fmaf(qd, kr[pp * HALO + qq], sc[pp * 7 + qq]);
    }

    float mx = sc[0];
    #pragma unroll
    for (int n = 1; n < 49; ++n) mx = fmaxf(mx, sc[n]);
    float ssum = 0.f;
    #pragma unroll
    for (int n = 0; n < 49; ++n) { sc[n] = expf(sc[n] - mx); ssum += sc[n]; }
    float inv = 1.0f / ssum;
    #pragma unroll
    for (int n = 0; n < 49; ++n) sc[n] *= inv;

    for (int d = 0; d < DH; ++d) {
        const float* vr = &Vlds[d * 196 + base];
        float o = 0.f;
        #pragma unroll
        for (int pp = 0; pp < 7; ++pp)
            #pragma unroll
            for (int qq = 0; qq < 7; ++qq)
                o = fmaf(sc[pp * 7 + qq], vr[pp * HALO + qq], o);
        out[(size_t)p * CCH + head * DH + d] = o;
    }
}

// ---------------------------------------------------------------------------
// LayerNorm over C (wave32 per pixel) fused with NHWC -> NCHW output write
// ---------------------------------------------------------------------------
__global__ __launch_bounds__(256) void ln_nhwc_to_nchw(
    const float* __restrict__ y, const float* __restrict__ g,
    const float* __restrict__ bt, float* __restrict__ out)
{
    int gt   = blockIdx.x * blockDim.x + threadIdx.x;
    int pix  = gt >> 5;
    int lane = threadIdx.x & 31;
    if (pix >= NPIX) return;

    v4f v = *(const v4f*)(y + (size_t)pix * CCH + lane * 4);
    float s  = v.x + v.y + v.z + v.w;
    float s2 = v.x * v.x + v.y * v.y + v.z * v.z + v.w * v.w;
    for (int off = 16; off > 0; off >>= 1) {
        s  += __shfl_xor(s,  off, 32);
        s2 += __shfl_xor(s2, off, 32);
    }
    float mean = s * (1.0f / 128.0f);
    float var  = s2 * (1.0f / 128.0f) - mean * mean;
    float rstd = rsqrtf(var + 1e-5f);

    int b  = pix >> 12;
    int ij = pix & 4095;
    #pragma unroll
    for (int t = 0; t < 4; ++t) {
        int c = lane * 4 + t;
        out[((size_t)(b * 128 + c)) * 4096 + ij] =
            (v[t] - mean) * rstd * g[c] + bt[c];
    }
}

// ---------------------------------------------------------------------------
extern "C" void kernel_launch(void* const* d_in, const int* in_sizes, int n_in,
                              void* d_out, int out_size, void* d_ws, size_t ws_size,
                              hipStream_t stream) {
    const float* x      = (const float*)d_in[0];
    const float* qkv_w  = (const float*)d_in[1];
    const float* qkv_b  = (const float*)d_in[2];
    const float* rpb    = (const float*)d_in[3];
    const float* proj_w = (const float*)d_in[4];
    const float* proj_b = (const float*)d_in[5];
    const float* ln_g   = (const float*)d_in[6];
    const float* ln_b   = (const float*)d_in[7];
    float* out = (float*)d_out;

    float* ws   = (float*)d_ws;
    float* buf0 = ws;                       // 8192*128
    float* buf1 = buf0 + NPIX * CCH;        // 8192*128
    float* qkvb = buf1 + NPIX * CCH;        // 8192*384
    float* attn = qkvb + NPIX * 384;        // 8192*128

    nchw_to_nhwc<<<8192, 256, 0, stream>>>(x, buf0);

    const float qscale = 0.17677669529663687f;   // 32^-0.5
    float* bin = buf0;
    float* bout = buf1;
    for (int l = 0; l < 2; ++l) {
        // QKV: waves = (8192/16)*(384/32) = 6144 -> 768 blocks of 256
        gemm_bias_wmma<<<768, 256, 0, stream>>>(
            bin, qkv_w + (size_t)l * 384 * 128, qkv_b + l * 384,
            qkvb, NPIX, 384, 128, qscale);
        // attention: 2 batches * 4 heads * 64 tiles
        natten_kernel<<<512, 64, 0, stream>>>(qkvb, rpb + l * NHEADS * 169, attn);
        // proj: waves = (8192/16)*(128/32) = 2048 -> 256 blocks of 256
        gemm_bias_wmma<<<256, 256, 0, stream>>>(
            attn, proj_w + (size_t)l * 128 * 128, proj_b + l * 128,
            bout, NPIX, 128, 0, 1.0f);
        float* t = bin; bin = bout; bout = t;
    }

    ln_nhwc_to_nchw<<<1024, 256, 0, stream>>>(bin, ln_g, ln_b, out);
}